// WeightedEmbeddingBag_16028817949384
// MI455X (gfx1250) — compile-verified
//
#include <hip/hip_runtime.h>

typedef __attribute__((ext_vector_type(2))) float v2f;
typedef __attribute__((ext_vector_type(8))) float v8f;

#define BATCH 4096
#define NITEM 200
#define NSEG  26
#define DIM   128

__global__ __launch_bounds__(256)
void weighted_embedding_bag_kernel(const int* __restrict__ input,      // [B, N] int32
                                   const float* __restrict__ psw,      // [B, N] f32
                                   const int* __restrict__ offsets,    // [B, M] int32 (sorted)
                                   const float* __restrict__ emb,      // [VOCAB, D] f32
                                   float* __restrict__ out)            // [B, M, D] f32
{
    const int b    = blockIdx.x;
    const int tid  = threadIdx.x;
    const int wave = tid >> 5;      // 0..7 -> D-tile
    const int lane = tid & 31;
    const int lh   = lane >> 4;     // lane half: 0 or 1
    const int l16  = lane & 15;

    __shared__ int   s_off[NSEG];
    __shared__ int   s_idx[NITEM];
    __shared__ float s_val[NITEM];
    __shared__ int   s_seg[NITEM];

    if (tid < NSEG) s_off[tid] = offsets[b * NSEG + tid];
    __syncthreads();

    // Stage indices/weights; compute segment id of each item:
    // seg[n] = smallest m with offsets[b,m] >= n  (item n lies in [off[m-1]+1, off[m]]).
    // Items beyond off[M-1] belong to no segment -> -1 -> A entry 0.
    for (int n = tid; n < NITEM; n += 256) {
        s_idx[n] = input[b * NITEM + n];
        s_val[n] = psw[b * NITEM + n];
        int seg = -1;
        #pragma unroll
        for (int m = 0; m < NSEG; ++m) {
            if (seg < 0 && s_off[m] >= n) seg = m;
        }
        s_seg[n] = seg;
    }
    __syncthreads();

    // score[b] = A(32x200) * E(200x128); this wave computes cols [dt*16, dt*16+16)
    // for both 16-row M-tiles, sharing the B fragment between the two WMMAs.
    const int dt  = wave;
    const int col = dt * 16 + l16;   // embedding column handled by this lane
    const int m0  = l16;             // output row, M-tile 0
    const int m1  = 16 + l16;        // output row, M-tile 1

    v8f acc0 = {0.f, 0.f, 0.f, 0.f, 0.f, 0.f, 0.f, 0.f};
    v8f acc1 = {0.f, 0.f, 0.f, 0.f, 0.f, 0.f, 0.f, 0.f};

    for (int kk = 0; kk < NITEM; kk += 4) {
        // 32-bit A/B fragment layout: VGPR0 holds K = kk + 2*laneHalf,
        // VGPR1 holds K = kk + 2*laneHalf + 1.
        const int ke = kk + 2 * lh;
        const int ko = ke + 1;

        const int i0 = s_idx[ke];
        const int i1 = s_idx[ko];
        v2f bfrag;
        bfrag.x = emb[(long)i0 * DIM + col];   // lanes 0-15: 64B contiguous (L2 hit)
        bfrag.y = emb[(long)i1 * DIM + col];

        const int   sg0 = s_seg[ke];
        const int   sg1 = s_seg[ko];
        const float w0  = s_val[ke];
        const float w1  = s_val[ko];

        v2f a0, a1;
        a0.x = (sg0 == m0) ? w0 : 0.f;
        a0.y = (sg1 == m0) ? w1 : 0.f;
        a1.x = (sg0 == m1) ? w0 : 0.f;
        a1.y = (sg1 == m1) ? w1 : 0.f;

        acc0 = __builtin_amdgcn_wmma_f32_16x16x4_f32(
            false, a0, false, bfrag, (short)0, acc0, false, false);
        acc1 = __builtin_amdgcn_wmma_f32_16x16x4_f32(
            false, a1, false, bfrag, (short)0, acc1, false, false);
    }

    // C/D layout: VGPR r -> row r (lanes 0-15) / row r+8 (lanes 16-31), col = lane&15.
    #pragma unroll
    for (int r = 0; r < 8; ++r) {
        const int mrow = r + lh * 8;
        out[((long)(b * NSEG + mrow)) * DIM + col] = acc0[r];
        const int m = 16 + mrow;
        if (m < NSEG)
            out[((long)(b * NSEG + m)) * DIM + col] = acc1[r];
    }
}

extern "C" void kernel_launch(void* const* d_in, const int* in_sizes, int n_in,
                              void* d_out, int out_size, void* d_ws, size_t ws_size,
                              hipStream_t stream) {
    (void)in_sizes; (void)n_in; (void)out_size; (void)d_ws; (void)ws_size;
    const int*   input   = (const int*)d_in[0];
    const float* psw     = (const float*)d_in[1];
    const int*   offsets = (const int*)d_in[2];
    const float* emb     = (const float*)d_in[3];
    float*       out     = (float*)d_out;

    weighted_embedding_bag_kernel<<<BATCH, 256, 0, stream>>>(input, psw, offsets, emb, out);
}